// GNNComponent_27693949125173
// MI455X (gfx1250) — compile-verified
//
#include <hip/hip_runtime.h>
#include <hip/hip_bf16.h>
#include <math.h>

// Problem constants (match reference)
#define NN 20000
#define EE 320000
#define ET (EE + NN)   // edges + self loops = 340000
#define FIN 256
#define DD 256
#define HH 4
#define CC 64
#define NEG_SLOPE 0.2f
#define LN_EPS 1e-5f

typedef float v2f __attribute__((ext_vector_type(2)));
typedef float v8f __attribute__((ext_vector_type(8)));

// ---------- monotonic float <-> uint key for atomic max over signed floats ----
__device__ __forceinline__ unsigned fkey(float f) {
    unsigned u = __float_as_uint(f);
    return (u & 0x80000000u) ? ~u : (u | 0x80000000u);
}
__device__ __forceinline__ float funkey(unsigned k) {
    unsigned u = (k & 0x80000000u) ? (k & 0x7FFFFFFFu) : ~k;
    return __uint_as_float(u);
}

// ---------- edge decode: first EE entries src, next EE dst; e>=EE => self loop
__device__ __forceinline__ void edge_sd(const long long* __restrict__ EI, int e,
                                        int& s, int& d) {
    if (e < EE) { s = (int)EI[e]; d = (int)EI[EE + e]; }
    else        { s = d = e - EE; }
}

// =====================  WMMA fp32 GEMM: C = A[N,256] x B[256,256] =============
// One wave computes one 16x16 tile using V_WMMA_F32_16X16X4_F32 (K-step 4).
// A 16x4 layout: K = k0 + 2*(lane>=16) + v, row = lane%16 (ISA 7.12.2).
// B 4x16 layout: same K mapping, col = lane%16.  C/D: row = r + 8*(lane>=16).
__global__ void wmma_gemm_f32_kernel(const float* __restrict__ A,
                                     const float* __restrict__ B,
                                     float* __restrict__ C, int nrows) {
    const int lane = threadIdx.x & 31;
    const int wave = threadIdx.x >> 5;                // 0..7
    const int i0 = blockIdx.x * 16;                   // row tile
    const int j0 = (blockIdx.y * 8 + wave) * 16;      // col tile (16 tiles total)
    if (i0 >= nrows) return;
    const int half = lane >> 4;                       // 0 or 1
    const int lid  = lane & 15;

    v8f c = {};
    const float* __restrict__ arow = A + (size_t)(i0 + lid) * DD;
    #pragma unroll 4
    for (int k0 = 0; k0 < DD; k0 += 4) {
        const int kb = k0 + half * 2;
        v2f a, b;
        a.x = arow[kb];
        a.y = arow[kb + 1];
        b.x = B[(size_t)kb       * DD + j0 + lid];
        b.y = B[(size_t)(kb + 1) * DD + j0 + lid];
        c = __builtin_amdgcn_wmma_f32_16x16x4_f32(
                /*neg_a=*/false, a, /*neg_b=*/false, b,
                /*c_mod=*/(short)0, c, /*reuse_a=*/false, /*reuse_b=*/false);
    }
    float* __restrict__ crow = C + (size_t)(i0 + half * 8) * DD + j0 + lid;
    #pragma unroll
    for (int r = 0; r < 8; ++r) crow[(size_t)r * DD] = c[r];
}

// =====================  init kernels (ws is poisoned -> must re-init) =========
__global__ void init_agg_kernel(float* __restrict__ out,
                                const float* __restrict__ bias) {
    out[(size_t)blockIdx.x * DD + threadIdx.x] = bias[threadIdx.x];
}
__global__ void init_stats_kernel(unsigned* __restrict__ mkey,
                                  float* __restrict__ z) {
    int t = blockIdx.x * blockDim.x + threadIdx.x;
    if (t < NN * HH) { mkey[t] = 0u; z[t] = 0.0f; }
}
__global__ void init_pool_kernel(float* __restrict__ acc,
                                 float* __restrict__ Zs,
                                 unsigned* __restrict__ gmk) {
    acc[threadIdx.x] = 0.0f;
    if (threadIdx.x == 0) { *Zs = 0.0f; *gmk = 0u; }
}

// =====================  per-(node,head) attention coefficients ================
__global__ void alpha_kernel(const float* __restrict__ h,
                             const float* __restrict__ a_src,
                             const float* __restrict__ a_dst,
                             float* __restrict__ as_, float* __restrict__ ad_) {
    int t = blockIdx.x * blockDim.x + threadIdx.x;   // n*HH + hh
    if (t >= NN * HH) return;
    int hh = t & (HH - 1), n = t >> 2;
    const float4* __restrict__ hv = (const float4*)(h + (size_t)n * DD + hh * CC);
    const float4* __restrict__ ws = (const float4*)(a_src + hh * CC);
    const float4* __restrict__ wd = (const float4*)(a_dst + hh * CC);
    float s = 0.f, d = 0.f;
    #pragma unroll
    for (int c = 0; c < CC / 4; ++c) {
        float4 v = hv[c], vs = ws[c], vd = wd[c];
        s += v.x * vs.x + v.y * vs.y + v.z * vs.z + v.w * vs.w;
        d += v.x * vd.x + v.y * vd.y + v.z * vd.z + v.w * vd.w;
    }
    as_[t] = s; ad_[t] = d;
}

// =====================  edge pass 1: leaky_relu + segment max =================
__global__ void edge_max_kernel(const float* __restrict__ as_,
                                const float* __restrict__ ad_,
                                const long long* __restrict__ EI,
                                unsigned* __restrict__ mkey,
                                float* __restrict__ ebuf) {
    int t = blockIdx.x * blockDim.x + threadIdx.x;
    if (t >= ET * HH) return;
    int hh = t & (HH - 1), e = t >> 2;
    int s, d; edge_sd(EI, e, s, d);
    float v = as_[s * HH + hh] + ad_[d * HH + hh];
    v = v > 0.f ? v : NEG_SLOPE * v;
    ebuf[t] = v;
    atomicMax(&mkey[d * HH + hh], fkey(v));
}

// =====================  edge pass 2: exp + segment sum ========================
__global__ void edge_exp_kernel(const long long* __restrict__ EI,
                                const unsigned* __restrict__ mkey,
                                float* __restrict__ ebuf,
                                float* __restrict__ z) {
    int t = blockIdx.x * blockDim.x + threadIdx.x;
    if (t >= ET * HH) return;
    int hh = t & (HH - 1), e = t >> 2;
    int s, d; edge_sd(EI, e, s, d);
    float m = funkey(mkey[d * HH + hh]);
    float w = expf(ebuf[t] - m);
    ebuf[t] = w;
    unsafeAtomicAdd(&z[d * HH + hh], w);
}

// =====================  edge pass 3: weighted scatter-add aggregation =========
// 64 threads per edge, float4 per thread (b128 gathers), 4 edges per block step.
__global__ void agg_kernel(const float* __restrict__ h,
                           const float* __restrict__ ebuf,
                           const float* __restrict__ z,
                           const long long* __restrict__ EI,
                           float* __restrict__ out) {
    const int sub = threadIdx.x >> 6;      // edge slot 0..3 within block
    const int q   = threadIdx.x & 63;      // quad index: dims [4q, 4q+4)
    const int hh  = q >> 4;                // head = (4q)/64
    for (int base = blockIdx.x * 4; base < ET; base += gridDim.x * 4) {
        int e = base + sub;
        if (e < ET) {
            int s, t; edge_sd(EI, e, s, t);
            float a = ebuf[e * HH + hh] / (z[t * HH + hh] + 1e-16f);
            float4 hv = *(const float4*)(h + (size_t)s * DD + q * 4);
            float* __restrict__ op = out + (size_t)t * DD + q * 4;
            unsafeAtomicAdd(op + 0, a * hv.x);
            unsafeAtomicAdd(op + 1, a * hv.y);
            unsafeAtomicAdd(op + 2, a * hv.z);
            unsafeAtomicAdd(op + 3, a * hv.w);
        }
    }
}

// =====================  LayerNorm (+residual) + ELU, one wave32 per row =======
__global__ void ln_elu_kernel(const float* __restrict__ in,
                              const float* __restrict__ g,
                              const float* __restrict__ be,
                              const float* __restrict__ resid, int useres,
                              float* __restrict__ out) {
    int row  = blockIdx.x * 8 + (threadIdx.x >> 5);
    int lane = threadIdx.x & 31;
    if (row >= NN) return;
    const float4* __restrict__ r4 = (const float4*)(in + (size_t)row * DD + lane * 8);
    float v[8];
    { float4 a = r4[0], b = r4[1];
      v[0]=a.x; v[1]=a.y; v[2]=a.z; v[3]=a.w; v[4]=b.x; v[5]=b.y; v[6]=b.z; v[7]=b.w; }
    float s = 0.f;
    #pragma unroll
    for (int i = 0; i < 8; ++i) s += v[i];
    #pragma unroll
    for (int off = 16; off > 0; off >>= 1) s += __shfl_xor(s, off);
    float mu = s * (1.0f / DD);
    float q = 0.f;
    #pragma unroll
    for (int i = 0; i < 8; ++i) { float dv = v[i] - mu; q += dv * dv; }
    #pragma unroll
    for (int off = 16; off > 0; off >>= 1) q += __shfl_xor(q, off);
    float rstd = rsqrtf(q * (1.0f / DD) + LN_EPS);
    #pragma unroll
    for (int i = 0; i < 8; ++i) {
        int dc = lane * 8 + i;
        float y = (v[i] - mu) * rstd * g[dc] + be[dc];
        if (useres) y += resid[(size_t)row * DD + dc];
        out[(size_t)row * DD + dc] = y > 0.f ? y : expm1f(y);  // ELU(alpha=1)
    }
}

// =====================  attention logits + global max =========================
__global__ void logits_kernel(const float* __restrict__ x2,
                              const float* __restrict__ aw,
                              const float* __restrict__ ab,
                              float* __restrict__ logits,
                              unsigned* __restrict__ gmk) {
    int row  = blockIdx.x * 8 + (threadIdx.x >> 5);
    int lane = threadIdx.x & 31;
    if (row >= NN) return;
    const float4* __restrict__ r4 = (const float4*)(x2 + (size_t)row * DD + lane * 8);
    const float4* __restrict__ w4 = (const float4*)(aw + lane * 8);
    float4 a = r4[0], b = r4[1], wa = w4[0], wb = w4[1];
    float s = a.x*wa.x + a.y*wa.y + a.z*wa.z + a.w*wa.w
            + b.x*wb.x + b.y*wb.y + b.z*wb.z + b.w*wb.w;
    #pragma unroll
    for (int off = 16; off > 0; off >>= 1) s += __shfl_xor(s, off);
    if (lane == 0) {
        float l = s + ab[0];
        logits[row] = l;
        atomicMax(gmk, fkey(l));
    }
}

// =====================  softmax-weighted pooling over nodes ===================
__global__ void pool_kernel(const float* __restrict__ x2,
                            const float* __restrict__ logits,
                            const unsigned* __restrict__ gmk,
                            float* __restrict__ acc,
                            float* __restrict__ Zs) {
    int d = threadIdx.x;
    float mx = funkey(*gmk);
    float local = 0.f, zloc = 0.f;
    for (int n = blockIdx.x; n < NN; n += gridDim.x) {
        float p = expf(logits[n] - mx);
        local += p * x2[(size_t)n * DD + d];
        if (d == 0) zloc += p;
    }
    unsafeAtomicAdd(&acc[d], local);
    if (d == 0) unsafeAtomicAdd(Zs, zloc);
}

__global__ void finalize_kernel(const float* __restrict__ acc,
                                const float* __restrict__ Zs,
                                float* __restrict__ out) {
    out[threadIdx.x] = acc[threadIdx.x] / (*Zs);
}

// ============================================================================
extern "C" void kernel_launch(void* const* d_in, const int* in_sizes, int n_in,
                              void* d_out, int out_size, void* d_ws, size_t ws_size,
                              hipStream_t stream) {
    const float*     x      = (const float*)d_in[0];
    const long long* EI     = (const long long*)d_in[1];
    const float*     W1     = (const float*)d_in[2];
    const float*     b1     = (const float*)d_in[3];
    const float*     a_s1   = (const float*)d_in[4];
    const float*     a_d1   = (const float*)d_in[5];
    const float*     W2     = (const float*)d_in[6];
    const float*     b2     = (const float*)d_in[7];
    const float*     a_s2   = (const float*)d_in[8];
    const float*     a_d2   = (const float*)d_in[9];
    const float*     g1     = (const float*)d_in[10];
    const float*     be1    = (const float*)d_in[11];
    const float*     g2     = (const float*)d_in[12];
    const float*     be2    = (const float*)d_in[13];
    const float*     attn_w = (const float*)d_in[14];
    const float*     attn_b = (const float*)d_in[15];
    float* outp = (float*)d_out;

    // -------- workspace layout (floats) --------
    float* P = (float*)d_ws;
    float* hbuf = P;                                  // N*D  (h1 / h2 / x2)
    float* aggo = hbuf + (size_t)NN * DD;             // N*D  (aggregated out)
    float* x1   = aggo + (size_t)NN * DD;             // N*D
    float* as_  = x1   + (size_t)NN * DD;             // N*H
    float* ad_  = as_  + (size_t)NN * HH;             // N*H
    unsigned* mk = (unsigned*)(ad_ + (size_t)NN * HH);// N*H
    float* z    = (float*)mk + (size_t)NN * HH;       // N*H
    float* ebuf = z + (size_t)NN * HH;                // ET*H
    float* logits = ebuf + (size_t)ET * HH;           // N
    float* acc  = logits + NN;                        // D
    float* Zs   = acc + DD;                           // 1
    unsigned* gmk = (unsigned*)(Zs + 1);              // 1

    const int nhBlocks   = (NN * HH + 255) / 256;     // 313
    const int edgeBlocks = (ET * HH + 255) / 256;     // 5313
    const dim3 gemmGrid(NN / 16, 2);                  // 1250 x 2, 8 waves/block
    const int rowBlocks  = NN / 8;                    // 2500 (wave-per-row kernels)

    // ================= Layer 1 =================
    init_agg_kernel  <<<NN, DD, 0, stream>>>(aggo, b1);
    init_stats_kernel<<<nhBlocks, 256, 0, stream>>>(mk, z);
    wmma_gemm_f32_kernel<<<gemmGrid, 256, 0, stream>>>(x, W1, hbuf, NN);
    alpha_kernel     <<<nhBlocks, 256, 0, stream>>>(hbuf, a_s1, a_d1, as_, ad_);
    edge_max_kernel  <<<edgeBlocks, 256, 0, stream>>>(as_, ad_, EI, mk, ebuf);
    edge_exp_kernel  <<<edgeBlocks, 256, 0, stream>>>(EI, mk, ebuf, z);
    agg_kernel       <<<4096, 256, 0, stream>>>(hbuf, ebuf, z, EI, aggo);
    ln_elu_kernel    <<<rowBlocks, 256, 0, stream>>>(aggo, g1, be1, nullptr, 0, x1);

    // ================= Layer 2 =================
    init_agg_kernel  <<<NN, DD, 0, stream>>>(aggo, b2);
    init_stats_kernel<<<nhBlocks, 256, 0, stream>>>(mk, z);
    wmma_gemm_f32_kernel<<<gemmGrid, 256, 0, stream>>>(x1, W2, hbuf, NN);
    alpha_kernel     <<<nhBlocks, 256, 0, stream>>>(hbuf, a_s2, a_d2, as_, ad_);
    edge_max_kernel  <<<edgeBlocks, 256, 0, stream>>>(as_, ad_, EI, mk, ebuf);
    edge_exp_kernel  <<<edgeBlocks, 256, 0, stream>>>(EI, mk, ebuf, z);
    agg_kernel       <<<4096, 256, 0, stream>>>(hbuf, ebuf, z, EI, aggo);
    // x2 = elu(LN(aggo) + x1)  -> reuse hbuf (h2 no longer needed)
    ln_elu_kernel    <<<rowBlocks, 256, 0, stream>>>(aggo, g2, be2, x1, 1, hbuf);

    // ================= Attention pooling =================
    init_pool_kernel <<<1, 256, 0, stream>>>(acc, Zs, gmk);
    logits_kernel    <<<rowBlocks, 256, 0, stream>>>(hbuf, attn_w, attn_b, logits, gmk);
    pool_kernel      <<<1024, 256, 0, stream>>>(hbuf, logits, gmk, acc, Zs);
    finalize_kernel  <<<1, DD, 0, stream>>>(acc, Zs, outp);
}